// BFPBatchNorm2d_custom_180388626599
// MI455X (gfx1250) — compile-verified
//
#include <hip/hip_runtime.h>

// BatchNorm2d forward (training stats), B=64 C=256 H=W=56, fp32.
// Bandwidth-bound: ~616 MB total traffic -> ~26us at 23.3 TB/s on MI455X.
// Three kernels: per-plane partial stats -> per-channel finalize -> apply.
// Deterministic (fixed-order reductions, no float atomics).

typedef float f4  __attribute__((ext_vector_type(4)));
typedef float v2f __attribute__((ext_vector_type(2)));
typedef float v8f __attribute__((ext_vector_type(8)));

#define BN_EPS 1e-5f
#define HW     3136   // 56*56
#define HW4    784    // HW/4 (float4 per plane)
#define BATCH  64
#define NCH    256
#define NPC    200704.0f  // BATCH*HW elements per channel

// ---------------------------------------------------------------------------
// Kernel 1: one block per (b,c) plane. Each thread accumulates sum / sumsq
// over float4 chunks, LDS-stage to wave 0, then reduce the 32 lane partials
// with V_WMMA_F32_16X16X4_F32:
//   A (16x4 f32) layout (ISA 7.12.2): lane m (0-15) holds K=0,1 in VGPR0,1;
//   lane m+16 holds K=2,3. We put the lane partial in both K slots and use an
//   all-ones B (layout-invariant), so D[m][n] = 2*(p(m) + p(m+16)).
//   D layout: lane n VGPR r = D[r][n] (lanes 16-31: D[r+8][n-16]); summing
//   VGPRs 0..7 and adding the lane+16 partner yields 2*total.
// ---------------------------------------------------------------------------
__global__ __launch_bounds__(256) void bn_stats_partial(
    const float* __restrict__ x, float* __restrict__ ps, float* __restrict__ pq) {
  const int bid = blockIdx.x;            // plane index = b*NCH + c (memory order)
  const int tid = threadIdx.x;
  const f4* x4 = (const f4*)x + (size_t)bid * HW4;

  float s = 0.f, q = 0.f;
  for (int j = tid; j < HW4; j += 256) {
    f4 v = x4[j];                        // global_load_b128, RT: warm L2 for pass 2
    s += v.x + v.y + v.z + v.w;
    q = fmaf(v.x, v.x, q); q = fmaf(v.y, v.y, q);
    q = fmaf(v.z, v.z, q); q = fmaf(v.w, v.w, q);
  }

  __shared__ float ls[256];
  __shared__ float lq[256];
  ls[tid] = s; lq[tid] = q;
  __syncthreads();

  if (tid < 32) {                        // wave 0, EXEC = 0xFFFFFFFF (WMMA req.)
    float sl = 0.f, ql = 0.f;
#pragma unroll
    for (int i = 0; i < 8; ++i) {        // conflict-free: stride-32 LDS reads
      sl += ls[tid + 32 * i];
      ql += lq[tid + 32 * i];
    }
#if __has_builtin(__builtin_amdgcn_wmma_f32_16x16x4_f32)
    v2f a_s = {sl, sl};
    v2f a_q = {ql, ql};
    v2f bones = {1.0f, 1.0f};
    v8f c0 = {};
    v8f ds = __builtin_amdgcn_wmma_f32_16x16x4_f32(
        false, a_s, false, bones, (short)0, c0, false, false);
    v8f dq = __builtin_amdgcn_wmma_f32_16x16x4_f32(
        false, a_q, false, bones, (short)0, c0, false, false);
    float rs = 0.f, rq = 0.f;
#pragma unroll
    for (int i = 0; i < 8; ++i) { rs += ds[i]; rq += dq[i]; }
    rs += __shfl_down(rs, 16);           // fold D rows 8..15 (lanes 16..31)
    rq += __shfl_down(rq, 16);
    rs *= 0.5f;                          // undo exact 2x from duplicated K slots
    rq *= 0.5f;
#else
    float rs = sl, rq = ql;
    for (int off = 16; off > 0; off >>= 1) {
      rs += __shfl_down(rs, off);
      rq += __shfl_down(rq, off);
    }
#endif
    if (tid == 0) {
      const int c = bid % NCH, b = bid / NCH;
      ps[c * BATCH + b] = rs;            // channel-contiguous for kernel 2
      pq[c * BATCH + b] = rq;
    }
  }
}

// ---------------------------------------------------------------------------
// Kernel 2: one block per channel, 64 threads reduce the 64 per-batch partials
// (fixed order -> deterministic), then compute scale/bias.
// ---------------------------------------------------------------------------
__global__ __launch_bounds__(64) void bn_finalize(
    const float* __restrict__ ps, const float* __restrict__ pq,
    const float* __restrict__ gamma, const float* __restrict__ beta,
    float* __restrict__ scale, float* __restrict__ bias) {
  const int c = blockIdx.x, tid = threadIdx.x;
  __shared__ float rs[64];
  __shared__ float rq[64];
  rs[tid] = ps[c * BATCH + tid];
  rq[tid] = pq[c * BATCH + tid];
  __syncthreads();
  for (int off = 32; off > 0; off >>= 1) {
    if (tid < off) { rs[tid] += rs[tid + off]; rq[tid] += rq[tid + off]; }
    __syncthreads();
  }
  if (tid == 0) {
    const float inv_n = 1.0f / NPC;
    const float mean = rs[0] * inv_n;
    const float var  = rq[0] * inv_n - mean * mean;  // biased variance
    const float sc   = gamma[c] * rsqrtf(var + BN_EPS);
    scale[c] = sc;
    bias[c]  = beta[c] - mean * sc;
  }
}

// ---------------------------------------------------------------------------
// Kernel 3: out = x*scale[c] + bias[c]. NT load on x (last use; keep L2 clean,
// still hits lines left by pass 1) and NT store on out (never re-read).
// ---------------------------------------------------------------------------
__global__ __launch_bounds__(256) void bn_apply(
    const float* __restrict__ x, const float* __restrict__ scale,
    const float* __restrict__ bias, float* __restrict__ out) {
  const int bid = blockIdx.x;
  const int c = bid % NCH;
  const float sc = scale[c];
  const float bs = bias[c];
  const f4* x4 = (const f4*)x + (size_t)bid * HW4;
  f4* o4 = (f4*)out + (size_t)bid * HW4;
  for (int j = threadIdx.x; j < HW4; j += 256) {
    f4 v = __builtin_nontemporal_load(&x4[j]);
    f4 o;
    o.x = fmaf(v.x, sc, bs);
    o.y = fmaf(v.y, sc, bs);
    o.z = fmaf(v.z, sc, bs);
    o.w = fmaf(v.w, sc, bs);
    __builtin_nontemporal_store(o, &o4[j]);
  }
}

extern "C" void kernel_launch(void* const* d_in, const int* in_sizes, int n_in,
                              void* d_out, int out_size, void* d_ws, size_t ws_size,
                              hipStream_t stream) {
  const float* x     = (const float*)d_in[0];
  const float* gamma = (const float*)d_in[1];
  const float* beta  = (const float*)d_in[2];
  float* out = (float*)d_out;

  float* ws    = (float*)d_ws;
  float* ps    = ws;                        // [NCH*BATCH] partial sums
  float* pq    = ws + NCH * BATCH;          // [NCH*BATCH] partial sumsq
  float* scale = ws + 2 * NCH * BATCH;      // [NCH]
  float* bias  = scale + NCH;               // [NCH]  (total 133,120 B of ws)

  const int planes = BATCH * NCH;           // 16384 blocks
  bn_stats_partial<<<dim3(planes), dim3(256), 0, stream>>>(x, ps, pq);
  bn_finalize<<<dim3(NCH), dim3(64), 0, stream>>>(ps, pq, gamma, beta, scale, bias);
  bn_apply<<<dim3(planes), dim3(256), 0, stream>>>(x, scale, bias, out);
}